// RelativePositionEncoding_8624294330880
// MI455X (gfx1250) — compile-verified
//
#include <hip/hip_runtime.h>

// ---------------------------------------------------------------------------
// RelativePositionEncoding for MI455X (gfx1250).
//
// Roofline: out = [1024,1024,128] fp32 = 512MB stores @ 23.3 TB/s => ~22us
// floor; compute is ~3 adds/element. Design = LDS-gather + wide NT stores.
//  * W (139x128 f32, 71KB) staged to LDS via CDNA5 async global->LDS copies
//    (ASYNCcnt / s_wait_asynccnt).
//  * Entity+sym terms pre-folded into a 12x128 table T with
//    v_wmma_f32_16x16x4_f32 (T = selector(12x8) x W[132:139], exact in f32).
//  * Hot loop: 1 pair per wave iteration, 3 ds_load_b128 + adds +
//    1 coalesced 512B nontemporal global_store_b128 per wave.
// ---------------------------------------------------------------------------

#define R_MAX   32
#define S_MAX   2
#define NDIM    128
#define NSEQ    1024
#define D_IN    139          // 2*(R_MAX+1)*2 + 1 + 2*(S_MAX+1)
#define O_TOK   66           // 2*(R_MAX+1)
#define O_ENT   132
#define O_SYM   133
#define JTILE   256          // pairs (j) per block
#define BLOCK   256          // 8 waves

typedef float v2f __attribute__((ext_vector_type(2)));
typedef float v4f __attribute__((ext_vector_type(4)));
typedef float v8f __attribute__((ext_vector_type(8)));

// Selector matrix A[m,k] for T = A x W[132 .. 139):
//   row m = e*6 + s (e = m>=6, s = m%6), col k: k==0 -> entity row W[132],
//   k==1..6 -> W[133+(k-1)], k==7 -> zero pad. Rows 12..15 pad to zero.
__device__ __forceinline__ float sel_aval(int m, int k) {
    if (m >= 12) return 0.0f;
    if (k == 0) return (m >= 6) ? 1.0f : 0.0f;
    int s = (m >= 6) ? (m - 6) : m;
    return (k >= 1 && k <= 6 && s == (k - 1)) ? 1.0f : 0.0f;
}

__global__ __launch_bounds__(BLOCK)
void rpe_kernel(const int* __restrict__ seq_index,
                const int* __restrict__ seq_color,
                const int* __restrict__ seq_sym,
                const int* __restrict__ seq_entity,
                const int* __restrict__ token_index,
                const float* __restrict__ W,
                float* __restrict__ out)
{
    __shared__ float sW[D_IN * NDIM];   // 71168 B
    __shared__ float sT[12 * NDIM];     //  6144 B   (fits: 320KB LDS/WGP)

    const int tid = threadIdx.x;

    // ---- Stage W into LDS: CDNA5 async global->LDS b128 copies ----------
    {
        unsigned ldsW = (unsigned)(unsigned long long)(void*)sW; // LDS byte offset
        const unsigned nchunks = (unsigned)(D_IN * NDIM * 4 / 16); // 4448
        for (unsigned k = (unsigned)tid; k < nchunks; k += BLOCK) {
            unsigned off = k * 16u;
            unsigned dst = ldsW + off;
            asm volatile("global_load_async_to_lds_b128 %0, %1, %2"
                         :
                         : "v"(dst), "v"(off), "s"(W)
                         : "memory");
        }
        asm volatile("s_wait_asynccnt 0" ::: "memory");
    }
    __syncthreads();

    // ---- Build T[12x128] = selector x W[132:139] with f32 WMMA (wave 0) --
    if (tid < 32) {
        const int lane = tid;
        const int mA   = lane & 15;            // A/B: lanes 0-15 and 16-31 both map M/N=0..15
        const int hi   = (lane >> 4) & 1;      // VGPR k-pairing: +0 (lanes 0-15) / +2 (16-31)
        const v8f zero = {0.f, 0.f, 0.f, 0.f, 0.f, 0.f, 0.f, 0.f};
        v8f acc[8];
        #pragma unroll
        for (int n = 0; n < 8; ++n) acc[n] = zero;

        #pragma unroll
        for (int kk = 0; kk < 2; ++kk) {       // K = 8 in two steps of 4
            const int kb = kk * 4 + hi * 2;
            v2f a;
            a.x = sel_aval(mA, kb + 0);
            a.y = sel_aval(mA, kb + 1);
            #pragma unroll
            for (int n = 0; n < 8; ++n) {      // 8 tiles of 16 columns
                const int col = n * 16 + mA;
                const int r0  = O_ENT + kb;    // W row for this k
                v2f b;
                b.x = (r0 + 0 < D_IN) ? sW[(r0 + 0) * NDIM + col] : 0.0f;
                b.y = (r0 + 1 < D_IN) ? sW[(r0 + 1) * NDIM + col] : 0.0f;
                acc[n] = __builtin_amdgcn_wmma_f32_16x16x4_f32(
                    false, a, false, b, (short)0, acc[n], false, false);
            }
        }
        // D layout: VGPR r -> M = r (lanes 0-15) / 8+r (lanes 16-31), N = lane%16
        #pragma unroll
        for (int n = 0; n < 8; ++n) {
            #pragma unroll
            for (int r = 0; r < 8; ++r) {
                const int mm = r + hi * 8;
                if (mm < 12) sT[mm * NDIM + n * 16 + mA] = acc[n][r];
            }
        }
    }
    __syncthreads();

    // ---- Hot loop: one (i,j) pair per wave iteration ---------------------
    const int i     = blockIdx.y;
    const int jbase = blockIdx.x * JTILE;
    const int wave  = __builtin_amdgcn_readfirstlane((int)threadIdx.x) >> 5; // uniform
    const int lane  = tid & 31;

    const int si = seq_index[i];
    const int ci = seq_color[i];
    const int yi = seq_sym[i];
    const int ei = seq_entity[i];
    const int ti = token_index[i];

    for (int jj = wave; jj < JTILE; jj += (BLOCK / 32)) {
        const int j  = jbase + jj;                 // wave-uniform -> SMEM loads
        const int sj = seq_index[j];
        const int cj = seq_color[j];
        const int yj = seq_sym[j];
        const int ej = seq_entity[j];
        const int tj = token_index[j];

        const bool bcol = (ci == cj);
        const bool bidx = (si == sj);
        const bool bent = (ei == ej);

        int dseq = si - sj;
        dseq = (dseq < -R_MAX ? -R_MAX : (dseq > R_MAX ? R_MAX : dseq)) + R_MAX;
        const int rseq = bcol ? dseq : (2 * R_MAX + 1);

        int dtok = ti - tj;
        dtok = (dtok < -R_MAX ? -R_MAX : (dtok > R_MAX ? R_MAX : dtok)) + R_MAX;
        const int rtok = (bcol && bidx) ? dtok : (2 * R_MAX + 1);

        int dsym = yi - yj;
        dsym = (dsym < -S_MAX ? -S_MAX : (dsym > S_MAX ? S_MAX : dsym)) + S_MAX;
        const int rT = bent ? (6 + dsym) : 5;      // T[e][s] folds entity term

        const v4f* pa = (const v4f*)(sW + rseq * NDIM);
        const v4f* pb = (const v4f*)(sW + (O_TOK + rtok) * NDIM);
        const v4f* pc = (const v4f*)(sT + rT * NDIM);

        const v4f o = pa[lane] + pb[lane] + pc[lane];

        // 512MB output, never re-read: non-temporal, wave = 512B coalesced
        v4f* po = (v4f*)(out + ((size_t)i * NSEQ + (size_t)j) * NDIM);
        __builtin_nontemporal_store(o, po + lane);
    }
}

extern "C" void kernel_launch(void* const* d_in, const int* in_sizes, int n_in,
                              void* d_out, int out_size, void* d_ws, size_t ws_size,
                              hipStream_t stream) {
    (void)in_sizes; (void)n_in; (void)d_ws; (void)ws_size; (void)out_size;
    const int*   seq_index   = (const int*)d_in[0];
    const int*   seq_color   = (const int*)d_in[1];
    const int*   seq_sym     = (const int*)d_in[2];
    const int*   seq_entity  = (const int*)d_in[3];
    const int*   token_index = (const int*)d_in[4];
    const float* W           = (const float*)d_in[5];
    float*       out         = (float*)d_out;

    dim3 grid(NSEQ / JTILE, NSEQ, 1);   // (4, 1024)
    dim3 block(BLOCK, 1, 1);
    rpe_kernel<<<grid, block, 0, stream>>>(seq_index, seq_color, seq_sym,
                                           seq_entity, token_index, W, out);
}